// SlicedWassersteinDistance_68968584839567
// MI455X (gfx1250) — compile-verified
//
#include <hip/hip_runtime.h>
#include <hip/hip_bf16.h>

// Sliced Wasserstein distance between persistence diagrams (MI455X / gfx1250).
//   Stage 1: WMMA fp32 16x16x4 projection of [X;Ydiag] / [Y;Xdiag] onto 100
//            directions; column-major per direction, +INF padded to 2^19 rows.
//   Stage 2: 200 segmented bitonic sorts:
//              - one fused LDS kernel fully sorts 8192-blocks (stages 2..8192)
//              - global merge passes, two strides fused per sweep (4 elem/thr)
//              - LDS-fused tail (strides <= 4096) per merge stage
//            LDS kernels move data with async global<->LDS (ASYNCcnt path).
//   Stage 3: deterministic two-stage reduction of sum |A - B| / K.
//
// Workspace: 200 * 2^19 * 4 B + 8 KB (~420 MB) in d_ws.

#define NPTS   200000            // N == M
#define NM     400000            // N + M
#define NDIR   100
#define PADN   524288            // 2^19 >= NM
#define NSEG   (2 * NDIR)
#define LDSN   8192              // elements per LDS sort block (32 KB)
#define RB1    2048

typedef __attribute__((ext_vector_type(2))) float v2f;
typedef __attribute__((ext_vector_type(8))) float v8f;

__device__ __forceinline__ void cmpex(float& x, float& y, bool asc) {
  const float lo = fminf(x, y);
  const float hi = fmaxf(x, y);
  x = asc ? lo : hi;
  y = asc ? hi : lo;
}

// ---------------------------------------------------------------------------
// Stage 1: projection via V_WMMA_F32_16X16X4_F32 (one wave per 16-row tile).
// ---------------------------------------------------------------------------
__global__ __launch_bounds__(32)
void swd_proj_wmma(const float* __restrict__ X, const float* __restrict__ Y,
                   float* __restrict__ buf) {
  const int lane   = threadIdx.x;
  const int i0     = blockIdx.x * 16;
  const int outSel = blockIdx.y;         // 0: rows=[X ; Ydiag]  1: rows=[Y ; Xdiag]
  const int r      = i0 + (lane & 15);

  v2f a = {0.f, 0.f};
  if (lane < 16 && r < NM) {
    const float* P = outSel ? Y : X;
    const float* Q = outSel ? X : Y;
    if (r < NPTS) {
      a.x = P[2 * r];
      a.y = P[2 * r + 1];
    } else {
      const int q = r - NPTS;
      const float t = 0.5f * (Q[2 * q] + Q[2 * q + 1]);
      a.x = t;  a.y = t;
    }
  }

  const float INF = __builtin_inff();
  const float PI  = 3.14159265358979323846f;

#pragma unroll
  for (int dt = 0; dt < 7; ++dt) {
    const int   col = dt * 16 + (lane & 15);
    const float th  = -0.5f * PI + (float)col * (PI / (float)NDIR);
    const float cc  = __cosf(th);
    const float ss  = __sinf(th);
    const float inv = 1.0f / (cc * cc + ss * ss);

    v2f b = {0.f, 0.f};
    if (lane < 16) { b.x = cc * inv; b.y = ss * inv; }

    v8f c = {};
    c = __builtin_amdgcn_wmma_f32_16x16x4_f32(false, a, false, b,
                                              (short)0, c, false, false);
    if (col < NDIR) {
      float* o = buf + (size_t)(outSel * NDIR + col) * PADN;
      const int half = (lane >> 4) * 8;
#pragma unroll
      for (int q = 0; q < 8; ++q) {
        const int row = i0 + half + q;
        o[row] = (row < NM) ? c[q] : INF;
      }
    }
  }
}

// ---------------------------------------------------------------------------
// Async global<->LDS helpers (gfx1250 ASYNCcnt path, 16 B per lane).
// ---------------------------------------------------------------------------
__device__ __forceinline__ void async_load_block(float* sm, const float* g) {
  const unsigned l0 = (unsigned)(uintptr_t)sm;   // low 32 bits == LDS byte offset
#pragma unroll
  for (int e = 0; e < 2; ++e) {
    const unsigned     lofs  = l0 + threadIdx.x * 16u + (unsigned)e * 16384u;
    const float* gaddr = g + threadIdx.x * 4 + e * 4096;
    asm volatile("global_load_async_to_lds_b128 %0, %1, off"
                 :: "v"(lofs), "v"(gaddr) : "memory");
  }
  asm volatile("s_wait_asynccnt 0x0" ::: "memory");
}

__device__ __forceinline__ void async_store_block(const float* sm, float* g) {
  const unsigned l0 = (unsigned)(uintptr_t)sm;
#pragma unroll
  for (int e = 0; e < 2; ++e) {
    const unsigned lofs  = l0 + threadIdx.x * 16u + (unsigned)e * 16384u;
    float*         gaddr = g + threadIdx.x * 4 + e * 4096;
    asm volatile("global_store_async_from_lds_b128 %0, %1, off"
                 :: "v"(gaddr), "v"(lofs) : "memory");
  }
  asm volatile("s_wait_asynccnt 0x0" ::: "memory");
}

// ---------------------------------------------------------------------------
// Stage 2a: fused local bitonic sort — completes ALL stages size=2..8192 for
// each 8192-element block in one memory sweep. Direction from global index.
// ---------------------------------------------------------------------------
__global__ __launch_bounds__(1024)
void swd_bitonic_local(float* __restrict__ buf) {
  __shared__ float sm[LDSN];
  const int seg  = blockIdx.x >> 6;            // PADN/LDSN == 64 blocks/segment
  const int boff = (blockIdx.x & 63) * LDSN;
  float* g = buf + (size_t)seg * PADN + boff;

  async_load_block(sm, g);
  __syncthreads();

  for (int size = 2; size <= LDSN; size <<= 1) {
    for (int stride = size >> 1; stride > 0; stride >>= 1) {
#pragma unroll 4
      for (int e = 0; e < 4; ++e) {
        const int p  = threadIdx.x + e * 1024;
        const int li = 2 * p - (p & (stride - 1));
        const bool asc = (((boff + li) & size) == 0);
        float x = sm[li], y = sm[li + stride];
        cmpex(x, y, asc);
        sm[li] = x; sm[li + stride] = y;
      }
      __syncthreads();
    }
  }

  async_store_block(sm, g);
}

// ---------------------------------------------------------------------------
// Stage 2b: global merge pass, ONE stride (stride >= 8192). 1 thread / pair.
// ---------------------------------------------------------------------------
__global__ __launch_bounds__(256)
void swd_bitonic_g1(float* __restrict__ buf, int size, int stride) {
  const int tid = blockIdx.x * 256 + threadIdx.x;
  const int seg = tid / (PADN / 2);
  const int p   = tid - seg * (PADN / 2);
  const int i   = 2 * p - (p & (stride - 1));
  float* s = buf + (size_t)seg * PADN;

  const bool asc = ((i & size) == 0);
  float x = s[i], y = s[i + stride];
  cmpex(x, y, asc);
  s[i] = x; s[i + stride] = y;
}

// ---------------------------------------------------------------------------
// Stage 2c: global merge pass, TWO strides fused (stride and stride/2, both
// >= 8192). Four elements per thread form a closed group under both strides;
// the ascending bit (size > stride) is shared by all four.
// ---------------------------------------------------------------------------
__global__ __launch_bounds__(256)
void swd_bitonic_g2(float* __restrict__ buf, int size, int stride) {
  const int tid = blockIdx.x * 256 + threadIdx.x;
  const int seg = tid / (PADN / 4);
  const int p   = tid - seg * (PADN / 4);
  const int sH  = stride >> 1;
  const int i0  = (p & (sH - 1)) | ((p & ~(sH - 1)) << 2);  // two zero bits inserted
  float* s = buf + (size_t)seg * PADN;

  const bool asc = ((i0 & size) == 0);
  float a = s[i0], b = s[i0 + sH], c = s[i0 + stride], d = s[i0 + stride + sH];
  cmpex(a, c, asc);  cmpex(b, d, asc);   // pass at `stride`
  cmpex(a, b, asc);  cmpex(c, d, asc);   // pass at `stride/2`
  s[i0] = a; s[i0 + sH] = b; s[i0 + stride] = c; s[i0 + stride + sH] = d;
}

// ---------------------------------------------------------------------------
// Stage 2d: LDS-fused merge tail — strides 4096..1 for one `size` stage.
// ---------------------------------------------------------------------------
__global__ __launch_bounds__(1024)
void swd_bitonic_lds(float* __restrict__ buf, int size) {
  __shared__ float sm[LDSN];
  const int seg  = blockIdx.x >> 6;
  const int boff = (blockIdx.x & 63) * LDSN;
  float* g = buf + (size_t)seg * PADN + boff;

  async_load_block(sm, g);
  __syncthreads();

  for (int stride = 4096; stride > 0; stride >>= 1) {
#pragma unroll 4
    for (int e = 0; e < 4; ++e) {
      const int p  = threadIdx.x + e * 1024;
      const int li = 2 * p - (p & (stride - 1));
      const bool asc = (((boff + li) & size) == 0);
      float x = sm[li], y = sm[li + stride];
      cmpex(x, y, asc);
      sm[li] = x; sm[li + stride] = y;
    }
    __syncthreads();
  }

  async_store_block(sm, g);
}

// ---------------------------------------------------------------------------
// Stage 3: deterministic two-stage reduction.
// ---------------------------------------------------------------------------
__global__ __launch_bounds__(256)
void swd_reduce1(const float* __restrict__ buf, float* __restrict__ partial) {
  __shared__ float sm[256];
  float acc = 0.f;
  const int total = NDIR * NM;
  for (int idx = blockIdx.x * 256 + threadIdx.x; idx < total; idx += RB1 * 256) {
    const int dir = idx / NM;
    const int i   = idx - dir * NM;
    acc += fabsf(buf[(size_t)dir * PADN + i] - buf[(size_t)(NDIR + dir) * PADN + i]);
  }
  sm[threadIdx.x] = acc;
  __syncthreads();
  for (int s = 128; s > 0; s >>= 1) {
    if (threadIdx.x < s) sm[threadIdx.x] += sm[threadIdx.x + s];
    __syncthreads();
  }
  if (threadIdx.x == 0) partial[blockIdx.x] = sm[0];
}

__global__ __launch_bounds__(256)
void swd_reduce2(const float* __restrict__ partial, float* __restrict__ out) {
  __shared__ float sm[256];
  float acc = 0.f;
  for (int i = threadIdx.x; i < RB1; i += 256) acc += partial[i];
  sm[threadIdx.x] = acc;
  __syncthreads();
  for (int s = 128; s > 0; s >>= 1) {
    if (threadIdx.x < s) sm[threadIdx.x] += sm[threadIdx.x + s];
    __syncthreads();
  }
  if (threadIdx.x == 0) out[0] = sm[0] * (1.0f / (float)NDIR);
}

// ---------------------------------------------------------------------------
extern "C" void kernel_launch(void* const* d_in, const int* in_sizes, int n_in,
                              void* d_out, int out_size, void* d_ws, size_t ws_size,
                              hipStream_t stream) {
  (void)in_sizes; (void)n_in; (void)out_size; (void)ws_size;
  const float* X = (const float*)d_in[0];
  const float* Y = (const float*)d_in[1];
  float* out     = (float*)d_out;
  float* buf     = (float*)d_ws;
  float* partial = buf + (size_t)NSEG * PADN;

  // 1) WMMA projections
  swd_proj_wmma<<<dim3(PADN / 16, 2), 32, 0, stream>>>(X, Y, buf);

  // 2) segmented bitonic sort
  const int ldsBlocks = NSEG * (PADN / LDSN);            // 12800
  const int g1Blocks  = (NSEG * (PADN / 2)) / 256;       // 1 pair/thread
  const int g2Blocks  = (NSEG * (PADN / 4)) / 256;       // 4 elems/thread

  swd_bitonic_local<<<ldsBlocks, 1024, 0, stream>>>(buf);  // all size<=8192

  for (int size = 2 * LDSN; size <= PADN; size <<= 1) {
    int stride = size >> 1;
    while (stride >= 8192) {
      if (stride >= 16384) {                               // fuse two passes
        swd_bitonic_g2<<<g2Blocks, 256, 0, stream>>>(buf, size, stride);
        stride >>= 2;
      } else {
        swd_bitonic_g1<<<g1Blocks, 256, 0, stream>>>(buf, size, stride);
        stride >>= 1;
      }
    }
    swd_bitonic_lds<<<ldsBlocks, 1024, 0, stream>>>(buf, size);  // strides<=4096
  }

  // 3) reduction
  swd_reduce1<<<RB1, 256, 0, stream>>>(buf, partial);
  swd_reduce2<<<1, 256, 0, stream>>>(partial, out);
}